// LatentSmoothnessTracker_12953621364922
// MI455X (gfx1250) — compile-verified
//
#include <hip/hip_runtime.h>

typedef __attribute__((ext_vector_type(2))) float v2f;
typedef __attribute__((ext_vector_type(8))) float v8f;

#define BSZ   64
#define TSZ   64
#define FSZ   8192
#define NPAIR 63
#define NWAVE 16
#define FCHUNK (FSZ / NWAVE)   // 512 f32 per wave

// One workgroup = one (batch, 16-row block). Each wave accumulates a partial
// 16x16 f32 Gram over its F-chunk via v_wmma_f32_16x16x4_f32 (A == B in
// registers because the A and B layouts are mutual transposes for this shape).
__global__ __launch_bounds__(512)
void gram16_kernel(const float* __restrict__ x, float* __restrict__ out) {
    __shared__ float lds[NWAVE * 256];   // per-wave Gram tiles (j*32 + lane)
    __shared__ float gram[256];          // reduced Gram tile

    const int tid  = threadIdx.x;
    const int wave = tid >> 5;
    const int lane = tid & 31;
    const int row  = lane & 15;          // M index (both lane halves cover M=0..15)
    const int koff = (lane >> 4) << 1;   // lanes 0-15 -> K0,K1 ; lanes 16-31 -> K2,K3

    const int blk = blockIdx.x % 5;
    const int b   = blockIdx.x / 5;
    const int t0  = (blk < 4) ? blk * 15 : 48;   // overlapped blocks cover pairs 0..62

    const float* rowp = x + (size_t)(b * TSZ + t0 + row) * FSZ + wave * FCHUNK + koff;

    v8f acc = {};
    #pragma unroll 8
    for (int f = 0; f < FCHUNK; f += 4) {
        v2f v = *(const v2f*)(rowp + f);   // global_load_b64, fully streaming
        // D = A * A^T + C  : Gram accumulate (A and B share registers)
        acc = __builtin_amdgcn_wmma_f32_16x16x4_f32(
            false, v, false, v, (short)0, acc, false, false);
    }

    // Spill per-wave Gram (8 VGPRs x 32 lanes) to LDS.
    #pragma unroll
    for (int j = 0; j < 8; ++j)
        lds[wave * 256 + j * 32 + lane] = acc[j];
    __syncthreads();

    // Reduce the 16 wave tiles elementwise.
    if (tid < 256) {
        float s = 0.0f;
        #pragma unroll
        for (int w = 0; w < NWAVE; ++w) s += lds[w * 256 + tid];
        gram[tid] = s;
    }
    __syncthreads();

    // D layout: VGPR j, lane l  ->  (M = j + (l>=16 ? 8 : 0), N = l & 15).
    if (tid < 15) {
        const int m = tid;                    // pair (t0+m, t0+m+1)
        auto G = [&](int mm, int nn) -> float {
            int j = mm & 7;
            int l = (mm < 8) ? nn : (nn + 16);
            return gram[j * 32 + l];
        };
        float na2 = G(m, m);
        float nb2 = G(m + 1, m + 1);
        float dot = G(m, m + 1);

        float denom = fmaxf(sqrtf(na2 * nb2), 1e-8f);
        float sim   = dot / denom;
        float cosd  = fmaxf(0.0f, 1.0f - sim);
        // mean((a-b)^2) = (|a|^2 + |b|^2 - 2 a.b) / F
        float mse   = (na2 + nb2 - 2.0f * dot) * (1.0f / (float)FSZ);

        int t = t0 + m;                       // t in [0, 62]
        out[b * NPAIR + t]                   = cosd;  // channel 0: cos_dist
        out[BSZ * NPAIR + b * NPAIR + t]     = mse;   // channel 1: mse
    }
}

extern "C" void kernel_launch(void* const* d_in, const int* in_sizes, int n_in,
                              void* d_out, int out_size, void* d_ws, size_t ws_size,
                              hipStream_t stream) {
    (void)in_sizes; (void)n_in; (void)d_ws; (void)ws_size; (void)out_size;
    const float* x = (const float*)d_in[0];
    float* out = (float*)d_out;
    dim3 grid(BSZ * 5);   // 64 batches x 5 row-blocks
    dim3 block(512);      // 16 waves
    gram16_kernel<<<grid, block, 0, stream>>>(x, out);
}